// EGAT_77790447665586
// MI455X (gfx1250) — compile-verified
//
#include <hip/hip_runtime.h>

typedef __attribute__((ext_vector_type(2))) float v2f;
typedef __attribute__((ext_vector_type(8))) float v8f;

#define N_NODES 100000
#define N_EDGES 1600000
#define IN_CH   64
#define OUT_CH  32

// ---------------------------------------------------------------------------
// Kernel 1: y[N,32] = x[N,64] @ W[32,64]^T via V_WMMA_F32_16X16X4_F32.
// One wave per 16x16 output tile. N_NODES = 6250*16 exactly -> no tails,
// EXEC is all-1s inside every executing wave (WMMA requirement).
// ---------------------------------------------------------------------------
__global__ __launch_bounds__(256) void egat_gemm_wmma(
    const float* __restrict__ x, const float* __restrict__ W,
    float* __restrict__ y, int numTiles) {
  const int lane = threadIdx.x & 31;
  const int wave = threadIdx.x >> 5;
  const int tile = blockIdx.x * 8 + wave;      // wave-uniform
  if (tile >= numTiles) return;                // uniform branch: EXEC stays full

  const int rowTile = tile >> 1;               // 6250 row tiles
  const int colTile = tile & 1;                // 2 col tiles (32 outputs)

  // A (16x4 f32): lane l -> M = l&15, K-pair base = (l>>4)*2
  // B (4x16 f32): lane l -> N = l&15, same K striping
  const int mn = lane & 15;
  const int kb = (lane >> 4) << 1;             // 0 or 2
  const float* __restrict__ aRow = x + (long)(rowTile * 16 + mn) * IN_CH + kb;
  const float* __restrict__ bRow = W + (long)(colTile * 16 + mn) * IN_CH + kb;

  v8f acc = {};
#pragma unroll
  for (int kk = 0; kk < IN_CH; kk += 4) {
    v2f a = *(const v2f*)(aRow + kk);          // 8B-aligned float2
    v2f b = *(const v2f*)(bRow + kk);
    acc = __builtin_amdgcn_wmma_f32_16x16x4_f32(
        /*neg_a=*/false, a, /*neg_b=*/false, b,
        /*c_mod=*/(short)0, acc, /*reuse_a=*/false, /*reuse_b=*/false);
  }

  // C/D layout: VGPR r -> M = r + (lane>=16 ? 8 : 0), N = lane&15
  const int Mbase = rowTile * 16 + ((lane >> 4) << 3);
  const int N     = colTile * 16 + (lane & 15);
#pragma unroll
  for (int r = 0; r < 8; ++r)
    y[(long)(Mbase + r) * OUT_CH + N] = acc[r];
}

// ---------------------------------------------------------------------------
// Kernel 2: zero the output accumulator (must run every launch; harness
// poisons d_out and replays the graph).
// ---------------------------------------------------------------------------
__global__ __launch_bounds__(256) void egat_zero(float* __restrict__ z, int n) {
  int i = blockIdx.x * blockDim.x + threadIdx.x;
  if (i < n) z[i] = 0.0f;
}

// ---------------------------------------------------------------------------
// Kernel 3: z[row[e]] += y[col[e]].  8 lanes per edge: float4 gather +
// 4 dword global_atomic_add_f32. y and z (12.8 MB each) are L2-resident
// (192 MB L2), so the random traffic never touches HBM.
// ---------------------------------------------------------------------------
__global__ __launch_bounds__(256) void egat_scatter(
    const long long* __restrict__ ei,  // [2, E] int64: row = ei[0,:], col = ei[1,:]
    const float* __restrict__ y, float* __restrict__ z) {
  long gid  = (long)blockIdx.x * blockDim.x + threadIdx.x;
  long e    = gid >> 3;
  int  part = (int)(gid & 7) << 2;     // channel group: 0,4,...,28
  if (e >= N_EDGES) return;

  long r = ei[e];                      // destination node
  long c = ei[(long)N_EDGES + e];      // source node

  const float4 v = *(const float4*)(y + c * OUT_CH + part);
  float* dst = z + r * OUT_CH + part;
  atomicAdd(dst + 0, v.x);
  atomicAdd(dst + 1, v.y);
  atomicAdd(dst + 2, v.z);
  atomicAdd(dst + 3, v.w);
}

extern "C" void kernel_launch(void* const* d_in, const int* in_sizes, int n_in,
                              void* d_out, int out_size, void* d_ws, size_t ws_size,
                              hipStream_t stream) {
  const float*     x  = (const float*)d_in[0];     // [N,64]
  const long long* ei = (const long long*)d_in[1]; // [2,E] int64
  // d_in[2] edge_attr, d_in[4] We, d_in[5] Wa: provably unused (softmax over
  // a size-1 axis makes alpha == 1, killing the whole attention branch).
  const float*     W  = (const float*)d_in[3];     // [32,64]
  float* z = (float*)d_out;                        // [N,32]
  float* y = (float*)d_ws;                         // [N,32] scratch (12.8 MB)

  // 1) zero output accumulator
  {
    int n = N_NODES * OUT_CH;
    egat_zero<<<(n + 255) / 256, 256, 0, stream>>>(z, n);
  }
  // 2) y = x @ W^T  (WMMA)
  {
    int numTiles = (N_NODES / 16) * 2;             // 12500
    int blocks   = (numTiles + 7) / 8;             // 8 waves per block
    egat_gemm_wmma<<<blocks, 256, 0, stream>>>(x, W, y, numTiles);
  }
  // 3) scatter-add per edge
  {
    long threads = (long)N_EDGES * 8;
    egat_scatter<<<(int)((threads + 255) / 256), 256, 0, stream>>>(ei, y, z);
  }
}